// Loca_901943132312
// MI455X (gfx1250) — compile-verified
//
#include <hip/hip_runtime.h>
#include <hip/hip_bf16.h>
#include <stdint.h>

// Problem constants (match reference)
#define ALPHA_F      0.95f
#define NCLS         32000
#define BLOCK        800                    // 25 wave32 waves
#define V4_PER_ROW   (NCLS / 4)             // 8000 float4 per row
#define V4_PER_THR   (V4_PER_ROW / BLOCK)   // 10 float4 per thread (exact)
#define NWAVES       (BLOCK / 32)           // 25

// Native clang vector: valid for __builtin_nontemporal_store and b128 ld/st.
typedef __attribute__((ext_vector_type(4))) float v4f;

// CDNA5 async global -> LDS copy, 16 bytes per lane per issue.
// GV addressing mode: vdst = LDS byte offset (VGPR), vaddr = 64-bit VA pair.
__device__ __forceinline__ void async_g2lds_b128(uint32_t lds_byte_off,
                                                 const void* gaddr) {
    asm volatile("global_load_async_to_lds_b128 %0, %1, off"
                 :: "v"(lds_byte_off), "v"((uint64_t)(uintptr_t)gaddr)
                 : "memory");
}

__device__ __forceinline__ void wait_asynccnt0() {
    asm volatile("s_wait_asynccnt 0x0" ::: "memory");
}

__global__ __launch_bounds__(BLOCK, 1)
void distill_label_smooth_kernel(const float* __restrict__ logits,
                                 const int*   __restrict__ labels,
                                 float*       __restrict__ out) {
    // 128000 B row buffer + small reduction scratch -> 2 workgroups / WGP (320 KB LDS)
    __shared__ __align__(16) float row_sh[NCLS];
    __shared__ float red_sh[32];
    __shared__ float S_sh;

    const int row = blockIdx.x;
    const int tid = threadIdx.x;
    const float* rowp = logits + (size_t)row * NCLS;
    float*       outp = out    + (size_t)row * NCLS;

    // LDS byte offset of row_sh: low 32 bits of the flat shared address are the
    // LDS offset (flat->LDS mapping truncates to addr[31:0]).
    const uint32_t row_sh_off = (uint32_t)(uintptr_t)(&row_sh[0]);

    // ---- Phase 0: stream the whole row into LDS via the async DMA path ----
    // 800 threads x 10 x B128 = 128 KB in flight, no VGPR return pressure.
#pragma unroll
    for (int k = 0; k < V4_PER_THR; ++k) {
        const int v4 = tid + k * BLOCK;                 // [0, 8000)
        async_g2lds_b128(row_sh_off + (uint32_t)(v4 * 16), rowp + v4 * 4);
    }
    wait_asynccnt0();   // our lanes' data has landed in our LDS region

    // ---- Phase 1: read back own chunks into registers, partial row sum ----
    v4f vals[V4_PER_THR];
    float lsum = 0.0f;
#pragma unroll
    for (int k = 0; k < V4_PER_THR; ++k) {
        const int v4 = tid + k * BLOCK;
        vals[k] = *reinterpret_cast<const v4f*>(&row_sh[v4 * 4]);
        lsum += (vals[k].x + vals[k].y) + (vals[k].z + vals[k].w);
    }

    // wave32 butterfly reduce
#pragma unroll
    for (int off = 16; off > 0; off >>= 1)
        lsum += __shfl_xor(lsum, off, 32);

    const int wave = tid >> 5;
    const int lane = tid & 31;
    if (lane == 0) red_sh[wave] = lsum;
    __syncthreads();

    if (wave == 0) {
        float w = (lane < NWAVES) ? red_sh[lane] : 0.0f;
#pragma unroll
        for (int off = 16; off > 0; off >>= 1)
            w += __shfl_xor(w, off, 32);
        if (lane == 0) S_sh = w;
    }
    __syncthreads();

    // ---- Phase 2: uniform scalars, then scale + single NT write ----
    const int   label = labels[row];        // uniform -> scalar load
    const float S     = S_sh;
    const float t     = row_sh[label];      // broadcast ds_load, no HBM traffic
    const float s     = ALPHA_F / (1.0f + S - 2.0f * t);
    const float corr  = 1.0f - s * S;       // out[label] = s*t + corr

    const int lv4   = label >> 2;
    const int lcomp = label & 3;

#pragma unroll
    for (int k = 0; k < V4_PER_THR; ++k) {
        const int v4 = tid + k * BLOCK;
        v4f v = vals[k];
        v.x *= s; v.y *= s; v.z *= s; v.w *= s;
        if (v4 == lv4) {                    // exactly one lane in the grid-row
            if      (lcomp == 0) v.x += corr;
            else if (lcomp == 1) v.y += corr;
            else if (lcomp == 2) v.z += corr;
            else                 v.w += corr;
        }
        // Output is never re-read: non-temporal store keeps L2 for input rows.
        __builtin_nontemporal_store(v, reinterpret_cast<v4f*>(outp + v4 * 4));
    }
}

extern "C" void kernel_launch(void* const* d_in, const int* in_sizes, int n_in,
                              void* d_out, int out_size, void* d_ws, size_t ws_size,
                              hipStream_t stream) {
    const float* logits = (const float*)d_in[0];
    const int*   labels = (const int*)d_in[1];
    float*       out    = (float*)d_out;
    const int batch = in_sizes[1];          // 4096 rows (one label per row)

    distill_label_smooth_kernel<<<batch, BLOCK, 0, stream>>>(logits, labels, out);
}